// WaveLetPooling_38843684225881
// MI455X (gfx1250) — compile-verified
//
#include <hip/hip_runtime.h>
#include <stdint.h>

// ---------------------------------------------------------------------------
// Haar 2x2 stride-2 subband decomposition, NHWC (8,512,512,64) fp32.
// Memory-bound: 1 GiB traffic -> ~46us floor at 23.3 TB/s.
// Data path: async global->LDS (ASYNCcnt) double-buffered pipeline,
// non-temporal b128 stores for the 4 subband outputs.
// ---------------------------------------------------------------------------

#define AS1 __attribute__((address_space(1)))
#define AS3 __attribute__((address_space(3)))

typedef float vfloat4 __attribute__((ext_vector_type(4)));
typedef int   v4i     __attribute__((__vector_size__(4 * sizeof(int))));

#define B_  8
#define H_  512
#define W_  512
#define C_  64
#define OH_ 256
#define OW_ 256
#define TILES_PER_ROW 16                     // 16 output pixels per tile
#define NUM_TILES (B_ * OH_ * TILES_PER_ROW) // 32768
#define CHUNKS 1024                          // 16B chunks per tile (2 rows x 512)
#define ROW_BYTES (W_ * C_ * 4)              // 131072 B per input row
#define SUBBAND_F4 (B_ * OH_ * OW_ * C_ / 4) // 8388608 float4 per subband

// ---- CDNA5 async global->LDS load (ASYNCcnt) ------------------------------
__device__ __forceinline__ void async_ld_b128(uint64_t gaddr, uint32_t ldsofs) {
#if __has_builtin(__builtin_amdgcn_global_load_async_to_lds_b128)
    __builtin_amdgcn_global_load_async_to_lds_b128(
        (AS1 v4i*)(uintptr_t)gaddr, (AS3 v4i*)ldsofs, /*offset=*/0, /*cpol=*/0);
#else
    // VDST = LDS byte address, VADDR = 64-bit global address, SADDR = off (GV mode)
    asm volatile("global_load_async_to_lds_b128 %0, %1, off"
                 :: "v"(ldsofs), "v"(gaddr) : "memory");
#endif
}

#if __has_builtin(__builtin_amdgcn_s_wait_asynccnt)
#define WAIT_ASYNC(n) __builtin_amdgcn_s_wait_asynccnt(n)
#else
#define WAIT_ASYNC(n) asm volatile("s_wait_asynccnt %0" :: "i"(n) : "memory")
#endif

__global__ __launch_bounds__(256) void haar_subbands_kernel(
    const float* __restrict__ x, float* __restrict__ out) {
    __shared__ vfloat4 smem[2][CHUNKS]; // 32 KB double buffer

    const int tid = threadIdx.x;
    const int stride = gridDim.x;
    const uint64_t xbase = (uint64_t)(uintptr_t)x;
    vfloat4* out4 = (vfloat4*)out;

    // Issue this thread's 4 async b128 loads for tile t into buffer `buf`.
    auto prefetch = [&](int t, int buf) {
        const int wt  = t & (TILES_PER_ROW - 1);
        const int rem = t >> 4;
        const int h   = rem & (OH_ - 1);
        const int b   = rem >> 8;
        // byte address of (b, 2h, wt*32, 0)
        const uint64_t base =
            xbase + ((uint64_t)((b * H_ + 2 * h) * W_ + wt * 32) * C_) * 4u;
#pragma unroll
        for (int i = 0; i < 4; ++i) {
            const int k   = tid + 256 * i;  // 16B chunk index in tile
            const int r   = k >> 9;         // input row within pair
            const int c16 = k & 511;        // 16B chunk within row segment
            const uint64_t g = base + (uint64_t)r * ROW_BYTES + (uint32_t)(c16 * 16);
            const uint32_t l = (uint32_t)(uintptr_t)&smem[buf][k];
            async_ld_b128(g, l);
        }
    };

    int t = (int)blockIdx.x;
    if (t < NUM_TILES) prefetch(t, 0);
    int cur = 0;

    for (; t < NUM_TILES; t += stride) {
        const int tn = t + stride;
        if (tn < NUM_TILES) {
            prefetch(tn, cur ^ 1);  // overlap next tile's fetch
            WAIT_ASYNC(4);          // in-order: current tile's 4 loads done
        } else {
            WAIT_ASYNC(0);
        }
        __syncthreads();            // all waves' data resident in LDS

        const int p  = tid >> 4;    // output pixel within tile [0,16)
        const int cg = tid & 15;    // float4 channel group [0,16)

        const vfloat4 a  = smem[cur][        (2 * p    ) * 16 + cg];
        const vfloat4 b4 = smem[cur][        (2 * p + 1) * 16 + cg];
        const vfloat4 c4 = smem[cur][512 +   (2 * p    ) * 16 + cg];
        const vfloat4 d4 = smem[cur][512 +   (2 * p + 1) * 16 + cg];

        const vfloat4 ab_s = a + b4;   // a+b
        const vfloat4 ab_d = b4 - a;   // b-a
        const vfloat4 cd_s = c4 + d4;  // c+d
        const vfloat4 cd_d = d4 - c4;  // d-c

        const vfloat4 ll = 0.5f * (ab_s + cd_s);
        const vfloat4 lh = 0.5f * (ab_d + cd_d);
        const vfloat4 hl = 0.5f * (cd_s - ab_s);
        const vfloat4 hh = 0.5f * (cd_d - ab_d);

        const int wt  = t & (TILES_PER_ROW - 1);
        const int rem = t >> 4;
        const int h   = rem & (OH_ - 1);
        const int b   = rem >> 8;
        const uint32_t opix = (uint32_t)((b * OH_ + h) * OW_ + wt * 16 + p);
        const uint32_t o    = opix * 16 + (uint32_t)cg;

        // outputs are never re-read: stream past the caches
        __builtin_nontemporal_store(ll, &out4[o]);
        __builtin_nontemporal_store(lh, &out4[o + 1u * SUBBAND_F4]);
        __builtin_nontemporal_store(hl, &out4[o + 2u * SUBBAND_F4]);
        __builtin_nontemporal_store(hh, &out4[o + 3u * SUBBAND_F4]);

        cur ^= 1;
        __syncthreads();            // protect buffer before next overwrite
    }
}

extern "C" void kernel_launch(void* const* d_in, const int* in_sizes, int n_in,
                              void* d_out, int out_size, void* d_ws, size_t ws_size,
                              hipStream_t stream) {
    (void)in_sizes; (void)n_in; (void)out_size; (void)d_ws; (void)ws_size;
    const float* x = (const float*)d_in[0];
    float* out = (float*)d_out;
    // 32768 tiles, 8 pipelined tiles per block
    haar_subbands_kernel<<<dim3(4096), dim3(256), 0, stream>>>(x, out);
}